// BaseSparseConn_9088150798841
// MI455X (gfx1250) — compile-verified
//
#include <hip/hip_runtime.h>
#include <cstdint>
#include <cstddef>

// ---------------------------------------------------------------------------
// COO SpMM: out(64,100000) = x(64,100000) @ conn + bias, conn given as COO of
// conn^T (rows = dst sorted ascending, cols = src), nnz = 1.6M.
//
// Memory-bound (AI ~0.4 FLOP/B). MI455X strategy:
//  * Pass 1: tiled transpose x -> x_t[IN][64] so each edge gather is one
//    contiguous 256B row; x_t (25.6MB) stays resident in the 192MB L2.
//  * Pass 2: row-partitioned WMMA scatter-accumulate. Block owns 64 output
//    rows (binary search in sorted rows for its edge range). Edge metadata is
//    staged into LDS with CDNA5 GLOBAL_LOAD_ASYNC_TO_LDS_B32 (per-lane
//    stride-8 gather of int64 low dwords, ASYNCcnt + s_wait_asynccnt).
//    Each of 4 waves owns a 16-wide batch slice; per chunk of 4 edges:
//    A = one-hot scatter matrix (16x4 f32, val_k at row rows[k]-winBase),
//    B = gathered activations (4x16), D accumulates via
//    V_WMMA_F32_16X16X4_F32 until the 16-row window slides -> plain
//    (atomic-free, deterministic) store flush.
//  * Pass 3: tiled transpose out_t -> out with bias add, non-temporal stores.
//
// Workspace: x_t (25.6MB) + out_t (25.6MB) = 51.2MB, assumed <= ws_size.
// ---------------------------------------------------------------------------

typedef __attribute__((ext_vector_type(2))) float v2f;
typedef __attribute__((ext_vector_type(8))) float v8f;

#define IN_FEATURES   100000
#define OUT_FEATURES  100000
#define BATCH         64
#define ROWS_PER_BLK  64
#define SLDS          1024   // edges staged in LDS per pass (12KB)

// ---------------- Pass 1: x (64, IN) -> x_t (IN, 64), tiled ----------------
__global__ void __launch_bounds__(256)
k_transpose_x(const float* __restrict__ x, float* __restrict__ xt) {
  __shared__ float tile[64][65];
  const int r0 = blockIdx.x * 64;
  const int t  = threadIdx.x;
  const int a  = t & 63, q = t >> 6;
  #pragma unroll 4
  for (int it = 0; it < 16; ++it) {
    const int b = q + 4 * it;
    const int r = r0 + a;
    tile[b][a] = (r < IN_FEATURES) ? x[(size_t)b * IN_FEATURES + r] : 0.f;
  }
  __syncthreads();
  #pragma unroll 4
  for (int it = 0; it < 16; ++it) {
    const int rl = q + 4 * it;
    const int r  = r0 + rl;
    if (r < IN_FEATURES) xt[(size_t)r * BATCH + a] = tile[a][rl];
  }
}

// ---------------- zero the accumulator (needed every launch) ----------------
__global__ void __launch_bounds__(256)
k_zero(float* __restrict__ p, size_t n) {
  size_t i = (size_t)blockIdx.x * blockDim.x + threadIdx.x;
  const size_t stride = (size_t)gridDim.x * blockDim.x;
  for (; i < n; i += stride) p[i] = 0.f;
}

// CDNA5 async copy: one b32 from global to LDS, tracked by ASYNCcnt.
// ldsOff is the wave-relative LDS byte offset = low 32 bits of the generic
// shared-aperture pointer (aperture lives entirely in addr[63:32], ISA 10.2).
__device__ __forceinline__ void async_g2l_b32(uint32_t ldsOff, uint64_t gaddr) {
  asm volatile("global_load_async_to_lds_b32 %0, %1, off"
               :: "v"(ldsOff), "v"(gaddr) : "memory");
}
__device__ __forceinline__ void wait_asynccnt0() {
  asm volatile("s_wait_asynccnt 0x0" ::: "memory");
}

// ---------------- Pass 2: WMMA scatter-accumulate SpMM ----------------
// Layout facts used (CDNA5 ISA 7.12.2, wave32):
//   A 16x4 f32 : lane l -> M = l%16 ; reg0 holds K = (l<16?0:2), reg1 K+1.
//   D 16x16 f32: reg j  -> M = j + (l<16?0:8), N = l%16.
//   B 4x16 f32 : assumed symmetric to A: lane l -> N = l%16 ;
//                reg0 holds K = (l<16?0:2), reg1 K+1.
__global__ void __launch_bounds__(128)
k_spmm_wmma(const float* __restrict__ xt, const float* __restrict__ vals,
            const long long* __restrict__ rows, const long long* __restrict__ cols,
            float* __restrict__ out_t, int nnz) {
  __shared__ __align__(16) int   s_row[SLDS];
  __shared__ __align__(16) int   s_col[SLDS];
  __shared__ __align__(16) float s_val[SLDS];
  __shared__ int s_range[2];

  const int r0 = blockIdx.x * ROWS_PER_BLK;
  const int t  = threadIdx.x;

  if (t == 0) {                       // binary-search this block's edge range
    const long long k0 = r0, k1 = r0 + ROWS_PER_BLK;
    int lo = 0, hi = nnz;
    while (lo < hi) { int mid = (lo + hi) >> 1; if (rows[mid] < k0) lo = mid + 1; else hi = mid; }
    const int e0 = lo;
    hi = nnz;
    while (lo < hi) { int mid = (lo + hi) >> 1; if (rows[mid] < k1) lo = mid + 1; else hi = mid; }
    s_range[0] = e0; s_range[1] = lo;
  }
  __syncthreads();
  const int eStart = s_range[0], eEnd = s_range[1];

  const int  lane   = t & 31;
  const int  wave   = t >> 5;            // 0..3 -> batch-column slice
  const int  n0     = wave * 16;
  const int  ml     = lane & 15;
  const bool hiHalf = lane >= 16;
  const int  rowMax = (r0 + ROWS_PER_BLK < OUT_FEATURES) ? (r0 + ROWS_PER_BLK) : OUT_FEATURES;

  const uint64_t rowsA = (uint64_t)(uintptr_t)rows;
  const uint64_t colsA = (uint64_t)(uintptr_t)cols;
  const uint64_t valsA = (uint64_t)(uintptr_t)vals;

  const v8f vzero = {};
  v8f acc = {};                          // 16x16 window accumulator (D/C)
  int winBase = r0;

// Flush: fast path = 8 unconditional stores off one divergent base pointer;
// slow path (window straddles block/feature end) = per-row guard.
#define FLUSH_WINDOW()                                                          \
  { if (winBase + 16 <= rowMax) {                                               \
      float* op_ = out_t + (size_t)(winBase + (hiHalf ? 8 : 0)) * BATCH + n0 + ml; \
      _Pragma("unroll")                                                         \
      for (int j = 0; j < 8; ++j) op_[(size_t)j * BATCH] = acc[j];              \
    } else {                                                                    \
      _Pragma("unroll")                                                         \
      for (int j = 0; j < 8; ++j) {                                             \
        const int rr_ = winBase + j + (hiHalf ? 8 : 0);                         \
        if (rr_ < rowMax) out_t[(size_t)rr_ * BATCH + n0 + ml] = acc[j];        \
      }                                                                         \
    } }

  for (int base = eStart; base < eEnd; base += SLDS) {
    int cnt = eEnd - base; if (cnt > SLDS) cnt = SLDS;
    const int cnt4 = (cnt + 3) & ~3;
    // async staging of edge metadata (int64 low-dword gather -> packed LDS)
    for (int i = t; i < cnt4; i += 128) {
      if (i < cnt) {
        const uint64_t e = (uint64_t)(base + i);
        async_g2l_b32((uint32_t)(uintptr_t)&s_row[i], rowsA + 8u * e);
        async_g2l_b32((uint32_t)(uintptr_t)&s_col[i], colsA + 8u * e);
        async_g2l_b32((uint32_t)(uintptr_t)&s_val[i], valsA + 4u * e);
      } else {                            // sentinel: zero contribution
        s_row[i] = 0x3FFFFFFF; s_col[i] = 0; s_val[i] = 0.f;
      }
    }
    wait_asynccnt0();
    __syncthreads();

    for (int i = 0; i < cnt4; i += 4) {
      // CDNA5 prefetch, two chunks ahead (global_prefetch_b8)
      if (i + 8 < cnt4) {
        const int cpf = __builtin_amdgcn_readfirstlane(s_col[i + 8 + (hiHalf ? 2 : 0)]);
        __builtin_prefetch(&xt[(size_t)cpf * BATCH + n0 + ml], 0, 1);
      }
      // chunk metadata via b128 LDS reads -> SGPRs, so control flow stays
      // scalar (EXEC == all 1s around WMMA, as the ISA requires)
      const int4 rr4 = *(const int4*)&s_row[i];
      const int4 cc4 = *(const int4*)&s_col[i];
      const int4 vv4 = *(const int4*)&s_val[i];
      const int   r0c = __builtin_amdgcn_readfirstlane(rr4.x);
      const int   r1c = __builtin_amdgcn_readfirstlane(rr4.y);
      const int   r2c = __builtin_amdgcn_readfirstlane(rr4.z);
      const int   r3c = __builtin_amdgcn_readfirstlane(rr4.w);
      const int   c0c = __builtin_amdgcn_readfirstlane(cc4.x);
      const int   c1c = __builtin_amdgcn_readfirstlane(cc4.y);
      const int   c2c = __builtin_amdgcn_readfirstlane(cc4.z);
      const int   c3c = __builtin_amdgcn_readfirstlane(cc4.w);
      const float v0c = __int_as_float(__builtin_amdgcn_readfirstlane(vv4.x));
      const float v1c = __int_as_float(__builtin_amdgcn_readfirstlane(vv4.y));
      const float v2c = __int_as_float(__builtin_amdgcn_readfirstlane(vv4.z));
      const float v3c = __int_as_float(__builtin_amdgcn_readfirstlane(vv4.w));

      // this lane's two K-slots (K = hiHalf ? {2,3} : {0,1})
      const int   rA = hiHalf ? r2c : r0c,  rB = hiHalf ? r3c : r1c;
      const int   cA = hiHalf ? c2c : c0c,  cB = hiHalf ? c3c : c1c;
      const float vA = hiHalf ? v2c : v0c;  const float vB = hiHalf ? v3c : v1c;

      // B operand depends only on columns: hoist the gathers above the
      // window-retry loop so they overlap the scalar window logic.
      v2f b;
      b.x = xt[(size_t)cA * BATCH + n0 + ml];          // 64B-coalesced gather
      b.y = xt[(size_t)cB * BATCH + n0 + ml];

      int rem = 0xF;
      while (rem) {
        // rows sorted -> first remaining edge has the minimum row
        const int rmin = (rem & 1) ? r0c : (rem & 2) ? r1c : (rem & 4) ? r2c : r3c;
        if (rmin >= winBase + 16) {       // window done: flush & slide
          FLUSH_WINDOW();
          acc = vzero;
          winBase = rmin;
        }
        const int lim = winBase + 16;
        const int in0 = (rem & 1) && (r0c < lim);
        const int in1 = (rem & 2) && (r1c < lim);
        const int in2 = (rem & 4) && (r2c < lim);
        const int in3 = (rem & 8) && (r3c < lim);
        const int iA  = hiHalf ? in2 : in0, iB = hiHalf ? in3 : in1;

        v2f a;
        a.x = (iA && (rA - winBase) == ml) ? vA : 0.f;  // one-hot scatter * val
        a.y = (iB && (rB - winBase) == ml) ? vB : 0.f;

        acc = __builtin_amdgcn_wmma_f32_16x16x4_f32(
            /*neg_a=*/false, a, /*neg_b=*/false, b,
            /*c_mod=*/(short)0, acc, /*reuse_a=*/false, /*reuse_b=*/false);

        rem &= ~(in0 | (in1 << 1) | (in2 << 2) | (in3 << 3));
      }
    }
    __syncthreads();                      // before re-staging LDS
  }
  FLUSH_WINDOW();                         // final partial window
#undef FLUSH_WINDOW
}

// ---------------- Pass 3: out_t (OUT,64) -> out (64,OUT) + bias ----------------
__global__ void __launch_bounds__(256)
k_finalize(const float* __restrict__ out_t, const float* __restrict__ bias,
           float* __restrict__ out) {
  __shared__ float tile[64][65];
  const int r0 = blockIdx.x * 64;
  const int t  = threadIdx.x;
  const int a  = t & 63, q = t >> 6;
  #pragma unroll 4
  for (int it = 0; it < 16; ++it) {
    const int rl = q + 4 * it;
    const int r  = r0 + rl;
    tile[rl][a] = (r < OUT_FEATURES) ? out_t[(size_t)r * BATCH + a] : 0.f;
  }
  __syncthreads();
  const int r = r0 + a;
  const float bz = (r < OUT_FEATURES) ? bias[r] : 0.f;
  #pragma unroll 4
  for (int it = 0; it < 16; ++it) {
    const int b = q + 4 * it;
    if (r < OUT_FEATURES)               // output read by nobody -> NT store
      __builtin_nontemporal_store(tile[a][b] + bz, &out[(size_t)b * OUT_FEATURES + r]);
  }
}

extern "C" void kernel_launch(void* const* d_in, const int* in_sizes, int n_in,
                              void* d_out, int out_size, void* d_ws, size_t ws_size,
                              hipStream_t stream) {
  const float*     x    = (const float*)d_in[0];
  const float*     vals = (const float*)d_in[1];
  const float*     bias = (const float*)d_in[2];
  const long long* rows = (const long long*)d_in[3];
  const long long* cols = (const long long*)d_in[4];
  float*           out  = (float*)d_out;
  const int        nnz  = in_sizes[1];

  float* xt    = (float*)d_ws;                              // IN*64 f32  (25.6MB)
  float* out_t = xt + (size_t)IN_FEATURES * BATCH;          // OUT*64 f32 (25.6MB)

  const int nblkIn  = (IN_FEATURES  + 63) / 64;
  const int nblkOut = (OUT_FEATURES + 63) / 64;

  k_transpose_x<<<nblkIn, 256, 0, stream>>>(x, xt);
  k_zero<<<2048, 256, 0, stream>>>(out_t, (size_t)OUT_FEATURES * BATCH);
  k_spmm_wmma<<<nblkOut, 128, 0, stream>>>(xt, vals, rows, cols, out_t, nnz);
  k_finalize<<<nblkOut, 256, 0, stream>>>(out_t, bias, out);
}